// CfConv_88880053223890
// MI455X (gfx1250) — compile-verified
//
#include <hip/hip_runtime.h>
#include <math.h>

// CfConv: out[b,n,m,f] = sum_k exp(-10*(||x_n - x_m|| - 0.1k)^2) * W_w[f,k] + W_b[f]
// B=4, N=768, D=20 RBFs, F=16. Matmul [16 pairs x 20] x [20 x 16] done as
// 5 chained V_WMMA_F32_16X16X4_F32 (exact fp32, K=4 each, 20 = 5*4).

typedef float v2f __attribute__((ext_vector_type(2)));
typedef float v8f __attribute__((ext_vector_type(8)));

#define B_DIM 4
#define N_DIM 768
#define D_NUM 20
#define F_OUT 16
#define GAMMA_F 10.0f

__global__ __launch_bounds__(256) void
cfconv_wmma_kernel(const float* __restrict__ coords,   // [B, N, 3]
                   const float* __restrict__ Ww,       // [F, D] row-major
                   const float* __restrict__ Wb,       // [F]
                   float* __restrict__ out)            // [B, N, N, F]
{
    const int lane = threadIdx.x & 31;
    const int wid  = threadIdx.x >> 5;
    const int row  = blockIdx.x * 8 + wid;     // row in [0, B*N)
    const int b    = row / N_DIM;
    const int n    = row - b * N_DIM;
    const int f    = lane & 15;                // feature column / pair row (M)
    const int half = lane >> 4;                // which K-half of the wave

    // B-matrix chunks (4x16 f32 each). ISA layout: lane = column N (=f),
    // VGPR0 = K0 (lanes 0-15) / K2 (lanes 16-31), VGPR1 = K1 / K3.
    // B[k][f] = Ww[f*D + k]; chunk c covers k = 4c .. 4c+3.
    v2f bm[5];
#pragma unroll
    for (int c = 0; c < 5; ++c) {
        const int k0 = 4 * c + 2 * half;
        bm[c].x = Ww[f * D_NUM + k0];
        bm[c].y = Ww[f * D_NUM + k0 + 1];
    }
    const float bias = Wb[f];

    // x_n, uniform across the wave
    const float xn0 = coords[(b * N_DIM + n) * 3 + 0];
    const float xn1 = coords[(b * N_DIM + n) * 3 + 1];
    const float xn2 = coords[(b * N_DIM + n) * 3 + 2];

    const size_t rowbase = (size_t)(b * N_DIM + n) * N_DIM * F_OUT;

    for (int m0 = 0; m0 < N_DIM; m0 += 16) {
        // A-matrix row M = lane%16 -> pair m = m0 + (lane%16); both wave
        // halves compute the same distance for their shared M row.
        const int m = m0 + f;
        const float dx = xn0 - coords[(b * N_DIM + m) * 3 + 0];
        const float dy = xn1 - coords[(b * N_DIM + m) * 3 + 1];
        const float dz = xn2 - coords[(b * N_DIM + m) * 3 + 2];
        const float d2 = dx * dx + dy * dy + dz * dz;
        const float d  = (d2 > 0.0f) ? sqrtf(d2) : 0.0f;

        // Accumulator pre-loaded with bias: D element (M, f) starts at Wb[f].
        v8f acc;
#pragma unroll
        for (int v = 0; v < 8; ++v) acc[v] = bias;

        // 5 chained K=4 fp32 WMMAs; each lane evaluates its 2 RBFs per chunk
        // (k = 4c + 2*half + {0,1}) -> 10 v_exp_f32 per lane per tile.
#pragma unroll
        for (int c = 0; c < 5; ++c) {
            const int k0 = 4 * c + 2 * half;
            const float t0 = d - 0.1f * (float)k0;
            const float t1 = d - 0.1f * (float)(k0 + 1);
            v2f a;
            a.x = __expf(-GAMMA_F * t0 * t0);
            a.y = __expf(-GAMMA_F * t1 * t1);
            acc = __builtin_amdgcn_wmma_f32_16x16x4_f32(
                /*neg_a=*/false, a, /*neg_b=*/false, bm[c],
                /*c_mod=*/(short)0, acc, /*reuse_a=*/false, /*reuse_b=*/false);
        }

        // D layout: VGPR v, lanes 0-15 -> M=v, lanes 16-31 -> M=v+8; N=f.
        // Streaming 151MB output -> non-temporal stores (don't pollute L2).
#pragma unroll
        for (int v = 0; v < 8; ++v) {
            const size_t idx =
                rowbase + (size_t)(m0 + v + 8 * half) * F_OUT + f;
            __builtin_nontemporal_store(acc[v], &out[idx]);
        }
    }
}

extern "C" void kernel_launch(void* const* d_in, const int* in_sizes, int n_in,
                              void* d_out, int out_size, void* d_ws, size_t ws_size,
                              hipStream_t stream) {
    (void)in_sizes; (void)n_in; (void)out_size; (void)d_ws; (void)ws_size;
    const float* coords = (const float*)d_in[0];   // [4, 768, 3]
    const float* Ww     = (const float*)d_in[1];   // [16, 20]
    const float* Wb     = (const float*)d_in[2];   // [16]
    float*       out    = (float*)d_out;           // [4, 768, 768, 16]

    const int rows   = B_DIM * N_DIM;              // 3072 waves, 1 row each
    const int blocks = rows / 8;                   // 256 threads = 8 waves
    cfconv_wmma_kernel<<<blocks, 256, 0, stream>>>(coords, Ww, Wb, out);
}